// PatchShift_79834852098334
// MI455X (gfx1250) — compile-verified
//
#include <hip/hip_runtime.h>
#include <cstdint>

// ---------------------------------------------------------------------------
// PatchShift (pattern 'D', RATIO=1, INV=false) == pure temporal gather:
//   out[bs, t, h, row, col, :] = x[bs, (t - S[row%4, col%4]) & 15, h, row, col, :]
// Shapes fixed by setup_inputs(): bs=16, T=16, H=12, N=196 (14x14), c=64, fp32.
// 0 FLOPs, 308 MB traffic -> HBM-bound, ~13.2 us floor at 23.3 TB/s.
// Streamed once each way -> non-temporal hints on both load and store.
// ---------------------------------------------------------------------------

#define NH        12
#define NPATCH    196
#define TLEN      16
#define C4        16                      // 64 floats = 16 x float4
#define F4_PER_B  (NH * NPATCH * C4)      // float4 stride of one (bs*T+t) slab = 37632

typedef __attribute__((ext_vector_type(4))) float v4f;

// Pattern 'D' shifts, +7 bias, packed 4 bits per (row%4*4 + col%4) entry.
//   idx 0..7  -> 0xCB06A987   idx 8..15 -> 0xF124DE39
__device__ __forceinline__ int shift_for(int n) {
  unsigned row = (unsigned)n / 14u;
  unsigned col = (unsigned)n - row * 14u;
  unsigned idx = ((row & 3u) << 2) | (col & 3u);
  unsigned word = (idx & 8u) ? 0xF124DE39u : 0xCB06A987u;
  return (int)((word >> ((idx & 7u) * 4u)) & 0xFu) - 7;
}

// ---------------------------------------------------------------------------
// Primary kernel: direct b128 permutation copy with NT cache hints.
// Each thread moves one float4; threadIdx.x = 16 n-positions x 16 float4s,
// so both the gather and the store are fully coalesced 256B channel rows.
// ---------------------------------------------------------------------------
__global__ __launch_bounds__(256) void patchshift_copy(
    const v4f* __restrict__ x, v4f* __restrict__ out) {
  const int v = threadIdx.x & (C4 - 1);
  const int n = blockIdx.x * 16 + (threadIdx.x >> 4);
  if (n >= NPATCH) return;
  const int h = blockIdx.y;
  const int b = blockIdx.z;              // b = bs*16 + t
  const int t = b & (TLEN - 1);
  const int s = shift_for(n);
  const int src_t = (t - s) & (TLEN - 1);
  const int i   = ((b * NH + h) * NPATCH + n) * C4 + v;   // out float4 index
  const int src = i + (src_t - t) * F4_PER_B;             // only t differs

  v4f val = __builtin_nontemporal_load(&x[src]);          // read-once: TH=NT
  __builtin_nontemporal_store(val, &out[i]);              // write-once: TH=NT
}

// ---------------------------------------------------------------------------
// CDNA5 async data-mover variant (syntax-validated on this toolchain):
// same permutation staged HBM -> LDS -> HBM via ASYNCcnt-tracked ops.
// Each thread owns a private 16B LDS slot, so no barrier is needed.
// Kept compiled as the gfx1250 async-path demonstration; not dispatched,
// because the direct copy avoids the per-wave s_wait_asynccnt serialization
// and the LDS round-trip while already sitting at the HBM roofline.
// ---------------------------------------------------------------------------
__global__ __launch_bounds__(256) void patchshift_async(
    const float* __restrict__ x, float* __restrict__ out) {
  __shared__ __align__(16) char smem[256 * 16];
  const int v = threadIdx.x & (C4 - 1);
  const int n = blockIdx.x * 16 + (threadIdx.x >> 4);
  if (n >= NPATCH) return;
  const int h = blockIdx.y;
  const int b = blockIdx.z;
  const int t = b & (TLEN - 1);
  const int s = shift_for(n);
  const int src_t = (t - s) & (TLEN - 1);
  const long long i   = ((long long)(b * NH + h) * NPATCH + n) * C4 + v;
  const long long src = i + (long long)(src_t - t) * F4_PER_B;

  const uint64_t srcA = (uint64_t)(uintptr_t)x   + (uint64_t)src * 16u;
  const uint64_t dstA = (uint64_t)(uintptr_t)out + (uint64_t)i   * 16u;
  const uint32_t lds  = (uint32_t)(uintptr_t)&smem[threadIdx.x * 16];

  asm volatile("global_load_async_to_lds_b128 %0, %1, off"
               :: "v"(lds), "v"(srcA) : "memory");
  asm volatile("s_wait_asynccnt 0x0" ::: "memory");
  asm volatile("global_store_async_from_lds_b128 %0, %1, off"
               :: "v"(dstA), "v"(lds) : "memory");
  // stores drain via the implicit wait-idle at s_endpgm
}

// ---------------------------------------------------------------------------
extern "C" void kernel_launch(void* const* d_in, const int* in_sizes, int n_in,
                              void* d_out, int out_size, void* d_ws, size_t ws_size,
                              hipStream_t stream) {
  (void)in_sizes; (void)n_in; (void)out_size; (void)d_ws; (void)ws_size;
  const v4f* x = (const v4f*)d_in[0];   // x: (256, 12, 196, 64) f32
  v4f* out = (v4f*)d_out;

  dim3 block(256, 1, 1);
  dim3 grid((NPATCH + 15) / 16,   // 13 n-tiles (last partially masked)
            NH,                   // 12 heads
            16 * TLEN);           // 256 = bs*T slabs
  patchshift_copy<<<grid, block, 0, stream>>>(x, out);
}